// ResHopf_5540507811920
// MI455X (gfx1250) — compile-verified
//
#include <hip/hip_runtime.h>
#include <math.h>
#include <stdint.h>

// ---------------------------------------------------------------------------
// ResHopf oscillator scan for MI455X (gfx1250, wave32).
//
// Shapes: T=2048 steps, B=16, U=512 -> 8192 independent oscillators.
// One thread per oscillator; each thread runs the serial recurrence and
// streams its trajectory out. Xr/Xi are prefetched DEPTH steps ahead via
// CDNA5 async global->LDS copies (ASYNCcnt), hiding HBM latency on the
// serial dependency chain. No barriers: each thread consumes only the LDS
// words it prefetched itself, and ASYNCcnt completes in-order per wave.
// ---------------------------------------------------------------------------

#define T_STEPS 2048
#define BATCH   16
#define UNITS   512
#define NOSC    (BATCH * UNITS)        /* 8192 oscillators                */
#define BLOCK   256                    /* 8 wave32 waves per workgroup    */
#define DEPTH   16                     /* async pipeline depth: 2*16=32   */
                                       /* outstanding <= ASYNCcnt max 63  */
#define STAGE_BYTES (2 * BLOCK * 4)    /* one stage = Xr slot + Xi slot   */

#define MU0f          1.0f
#define BETA1f        1.0f
#define INPUT_SCALERf 5.0f
#define MIN_OMEGAf    0.5f
#define MAX_OMEGAf    20.0f
#define DTf           0.01f
#define TWO_PIf       6.2831853071795864769f

// CDNA5 async copy: copies one float from global address gaddr to the LDS byte address, tracked by ASYNCcnt.
// Inline asm (not the clang builtin) per the bridge doc: portable across
// ROCm 7.2 and amdgpu-toolchain, and avoids addrspace-typed builtin args.
__device__ __forceinline__ void async_load_f32_to_lds(const float* gaddr,
                                                      uint32_t lds_byte_addr) {
    asm volatile("global_load_async_to_lds_b32 %0, %1, off"
                 :: "v"(lds_byte_addr), "v"(gaddr)
                 : "memory");
}

// Split dependency-counter waits (CDNA5 replaces vmcnt/lgkmcnt).
#define S_WAIT_ASYNCCNT(n) asm volatile("s_wait_asynccnt %0" :: "n"(n) : "memory")
#define S_WAIT_DSCNT0()    asm volatile("s_wait_dscnt 0" ::: "memory")

__global__ __launch_bounds__(BLOCK)
void reshopf_scan_kernel(const float* __restrict__ Xr,
                         const float* __restrict__ Xi,
                         const float* __restrict__ r0,
                         const float* __restrict__ phi0,
                         const float* __restrict__ omegas,
                         float* __restrict__ out) {
    // Stage s occupies buf[s*2*BLOCK .. s*2*BLOCK + 2*BLOCK):
    //   [0,BLOCK)      = Xr values for this stage (one per thread)
    //   [BLOCK,2BLOCK) = Xi values
    __shared__ float buf[DEPTH * 2 * BLOCK];   // 32 KB of 320 KB WGP LDS

    const int tid = threadIdx.x;
    const int j   = blockIdx.x * BLOCK + tid;  // flat oscillator id in [0,8192)
    const int u   = j & (UNITS - 1);           // unit index for omegas[1,U]

    // Flat LDS addresses have the LDS byte offset in the low 32 bits
    // (aperture bits live in [63:32]), which is exactly what the async
    // instruction's VDST expects (wave-LDS-relative byte address).
    const uint32_t lds0 = (uint32_t)(uintptr_t)(&buf[tid]);

    const float* gr = Xr + j;                  // stride NOSC floats per step
    const float* gi = Xi + j;

    // ---- prologue: fill the DEPTH-stage async pipeline -------------------
#pragma unroll
    for (int s = 0; s < DEPTH; ++s) {
        async_load_f32_to_lds(gr + (size_t)s * NOSC,
                              lds0 + s * STAGE_BYTES);
        async_load_f32_to_lds(gi + (size_t)s * NOSC,
                              lds0 + s * STAGE_BYTES + BLOCK * 4);
    }

    // ---- per-oscillator constants & state --------------------------------
    float r   = r0[j];
    float phi = phi0[j];
    const float sig = 1.0f / (1.0f + __expf(-omegas[u]));          // sigmoid
    const float om  = (sig * (MAX_OMEGAf - MIN_OMEGAf) + MIN_OMEGAf) * TWO_PIf;

    float sph, cph;                 // sin/cos of current phi, carried across
    __sincosf(phi, &sph, &cph);     // steps so only ONE sincos per iteration

    float2* __restrict__ zout = (float2*)out;                   // interleaved
    float*  __restrict__ rf   = out + (size_t)2 * T_STEPS * NOSC; // re/im c64
    float*  __restrict__ pf   = rf + NOSC;

    // ---- main scan --------------------------------------------------------
    for (int t = 0; t < T_STEPS; ++t) {
        // Oldest stage resident: async ops complete in-order per wave, so
        // count <= 2*(DEPTH-1) means both loads of stage t have landed.
        S_WAIT_ASYNCCNT(2 * (DEPTH - 1));

        const int slot = t & (DEPTH - 1);
        const float xr = buf[slot * (2 * BLOCK) + tid];
        const float xi = buf[slot * (2 * BLOCK) + BLOCK + tid];
        S_WAIT_DSCNT0();   // LDS reads retired before the slot is refilled

        const int tp = t + DEPTH;  // uniform across the wave -> no EXEC issue
        if (tp < T_STEPS) {
            async_load_f32_to_lds(gr + (size_t)tp * NOSC,
                                  lds0 + slot * STAGE_BYTES);
            async_load_f32_to_lds(gi + (size_t)tp * NOSC,
                                  lds0 + slot * STAGE_BYTES + BLOCK * 4);
        }

        const float in_r   = INPUT_SCALERf * xr * cph;   // uses cos(phi_old)
        const float in_phi = INPUT_SCALERf * xi * sph;   // uses sin(phi_old)
        r   = r + ((MU0f - BETA1f * r * r) * r + in_r) * DTf;
        phi = phi + (om - in_phi) * DTf;

        __sincosf(phi, &sph, &cph);                      // v_sin/v_cos (TRANS)

        // z[t,b,u] as complex64 = interleaved {re, im}: 8B per thread,
        // 256B contiguous per wave32 -> coalesced global_store_b64.
        zout[(size_t)t * NOSC + j] = make_float2(r * cph, r * sph);
    }

    // final state (r_f, phi_f) appended after z in the flat output
    rf[j] = r;
    pf[j] = phi;
}

extern "C" void kernel_launch(void* const* d_in, const int* in_sizes, int n_in,
                              void* d_out, int out_size, void* d_ws, size_t ws_size,
                              hipStream_t stream) {
    (void)in_sizes; (void)n_in; (void)out_size; (void)d_ws; (void)ws_size;
    const float* Xr     = (const float*)d_in[0];
    const float* Xi     = (const float*)d_in[1];
    const float* r0     = (const float*)d_in[2];
    const float* phi0   = (const float*)d_in[3];
    const float* omegas = (const float*)d_in[4];

    dim3 grid(NOSC / BLOCK);   // 32 workgroups x 256 threads = 8192 threads
    reshopf_scan_kernel<<<grid, BLOCK, 0, stream>>>(
        Xr, Xi, r0, phi0, omegas, (float*)d_out);
}